// CSANLayer_41223096107303
// MI455X (gfx1250) — compile-verified
//
#include <hip/hip_runtime.h>

// ---------------------------------------------------------------------------
// CSAN layer for MI455X (gfx1250, wave32).
//  Phase 1: fs = feat @ W_fc^T via V_WMMA_F32_16X16X4_F32 (one wave / 16x16 tile)
//  Phase 2: per-etype gather/scatter attention — memory bound; atomics land in
//           the 192MB L2 (all scatter targets are <52MB), float4 gathers.
// ---------------------------------------------------------------------------

#define EMB 256
#define HH  4
#define DD  32
#define HD  128   // H*D

typedef __attribute__((ext_vector_type(2))) float v2f;
typedef __attribute__((ext_vector_type(8))) float v8f;

// ---------------- Phase 1: projection GEMM (fp32 WMMA) ---------------------
// Grid: one block (8 waves) per 16-row tile; wave w handles column tile w (8
// column tiles cover HD=128). K loop: 256/4 = 64 wmma ops per tile.
__global__ __launch_bounds__(256) void proj_kernel(const float* __restrict__ feat,
                                                   const float* __restrict__ Wfc,
                                                   float* __restrict__ fs,
                                                   int nRowTiles) {
  int wave    = blockIdx.x * (blockDim.x >> 5) + (threadIdx.x >> 5);
  int rowTile = wave >> 3;
  int colTile = wave & 7;
  if (rowTile >= nRowTiles) return;          // uniform per-wave (EXEC stays all-1)

  int lane = threadIdx.x & 31;
  int m    = lane & 15;                      // M (A) / N (B) index
  int kh   = lane >> 4;                      // which K-half this lane holds

  // A: feat rows (16xK). ISA A-layout 16x4 f32: VGPR0={K0 | K2}, VGPR1={K1 | K3}
  //    -> lane (m, kh) loads consecutive pair at K = k0 + 2*kh. float2 load.
  // B: B[k][n] = Wfc[n][k]  (B is 4x16, same VGPR pattern as A mirrored on N).
  const float* aptr = feat + (size_t)(rowTile * 16 + m) * EMB + 2 * kh;
  const float* bptr = Wfc  + (size_t)(colTile * 16 + m) * EMB + 2 * kh;

  v8f acc = {};
#pragma unroll 8
  for (int k0 = 0; k0 < EMB; k0 += 4) {
    v2f a = *(const v2f*)(aptr + k0);
    v2f b = *(const v2f*)(bptr + k0);
    acc = __builtin_amdgcn_wmma_f32_16x16x4_f32(
        /*neg_a=*/false, a, /*neg_b=*/false, b,
        /*c_mod=*/(short)0, acc, /*reuse_a=*/false, /*reuse_b=*/false);
  }

  // C/D layout: VGPR r -> M = r + 8*kh, N = lane&15
  float* o = fs + (size_t)(rowTile * 16 + 8 * kh) * HD + colTile * 16 + m;
#pragma unroll
  for (int r = 0; r < 8; ++r) o[(size_t)r * HD] = acc[r];
}

// ---------------- helpers --------------------------------------------------
__global__ void zero_kernel(float* __restrict__ p, long n) {
  long i = blockIdx.x * (long)blockDim.x + threadIdx.x;
  long s = gridDim.x * (long)blockDim.x;
  for (; i < n; i += s) p[i] = 0.0f;
}

// order-preserving float<->uint mapping for atomicMax-based segment max
__device__ __forceinline__ unsigned fenc(float f) {
  unsigned u = __float_as_uint(f);
  return (u & 0x80000000u) ? ~u : (u | 0x80000000u);
}
__device__ __forceinline__ float fdec(unsigned k) {
  return (k & 0x80000000u) ? __uint_as_float(k ^ 0x80000000u)
                           : __uint_as_float(~k);
}

// ---------------- Phase 2 kernels ------------------------------------------
// 32 lanes per edge, float4 (global_load_b128) gather of fs[src], atomic
// scatter-add into acc[dst] (L2-resident, 51MB). Lane 0 bumps degree.
__global__ __launch_bounds__(256) void scatter_sum_kernel(
    const float* __restrict__ fs, const int* __restrict__ src,
    const int* __restrict__ dst, float* __restrict__ acc,
    float* __restrict__ deg, int E) {
  long t = blockIdx.x * 256L + threadIdx.x;
  long e = t >> 5;
  if (e >= E) return;
  int lane = (int)(t & 31);
  int s = src[e], d = dst[e];
  const float4 v = *(const float4*)(fs + (size_t)s * HD + lane * 4);
  float* ap = acc + (size_t)d * HD + lane * 4;
  atomicAdd(ap + 0, v.x); atomicAdd(ap + 1, v.y);
  atomicAdd(ap + 2, v.z); atomicAdd(ap + 3, v.w);
  if (lane == 0) atomicAdd(deg + d, 1.0f);
}

// one thread per (node, head): gating dots + el/er; also (re)inits the
// per-(node,head) softmax reduction slots for this etype.
__global__ __launch_bounds__(256) void node_kernel(
    const float* __restrict__ fs, const float* __restrict__ acc,
    const float* __restrict__ deg,
    const float* __restrict__ al, const float* __restrict__ ar,
    const float* __restrict__ tal, const float* __restrict__ tar,
    float* __restrict__ el, float* __restrict__ er, float* __restrict__ gate,
    unsigned* __restrict__ emaxE, float* __restrict__ denom, int Nn) {
  int t = blockIdx.x * 256 + threadIdx.x;
  if (t >= Nn * HH) return;
  int n = t >> 2, h = t & 3;
  const float* f = fs  + (size_t)n * HD + h * DD;
  const float* a = acc + (size_t)n * HD + h * DD;
  float invdeg = 1.0f / fmaxf(deg[n], 1.0f);
  float elt = 0.f, ert = 0.f, e_l = 0.f, e_r = 0.f;
#pragma unroll
  for (int dd = 0; dd < DD; ++dd) {
    float fv = f[dd];
    elt += fv    * tal[h * DD + dd];
    ert += a[dd] * tar[h * DD + dd];
    e_l += fv    * al[h * DD + dd];
    e_r += fv    * ar[h * DD + dd];
  }
  ert *= invdeg;                                  // <mean_dst, type_attn_r>
  float g = 1.0f / (1.0f + __expf(-(elt + ert))); // sigmoid gate
  el[t] = e_l; er[t] = e_r; gate[t] = g;
  emaxE[t] = 0x007FFFFFu;   // fenc(-inf)
  denom[t] = 0.0f;
}

// one thread per (edge, head): score + leaky_relu + atomic segment-max
__global__ __launch_bounds__(256) void score_kernel(
    const int* __restrict__ src, const int* __restrict__ dst,
    const float* __restrict__ el, const float* __restrict__ er,
    const float* __restrict__ gate, float* __restrict__ escore,
    unsigned* __restrict__ emaxE, int E) {
  long t = blockIdx.x * 256L + threadIdx.x;
  if (t >= (long)E * HH) return;
  long e = t >> 2; int h = (int)(t & 3);
  int s = src[e], d = dst[e];
  float v = gate[d * HH + h] * (el[s * HH + h] + er[d * HH + h]);
  v = (v >= 0.0f) ? v : 0.2f * v;                 // leaky_relu(0.2)
  escore[t] = v;
  atomicMax(emaxE + (size_t)d * HH + h, fenc(v));
}

// exp(e - max) + atomic segment-sum of denominators
__global__ __launch_bounds__(256) void exp_kernel(
    const int* __restrict__ dst, float* __restrict__ escore,
    const unsigned* __restrict__ emaxE, float* __restrict__ denom, int E) {
  long t = blockIdx.x * 256L + threadIdx.x;
  if (t >= (long)E * HH) return;
  long e = t >> 2; int h = (int)(t & 3);
  int d = dst[e];
  float m  = fdec(emaxE[(size_t)d * HH + h]);
  float ex = __expf(escore[t] - m);
  escore[t] = ex;
  atomicAdd(denom + (size_t)d * HH + h, ex);
}

// weighted aggregation: out[dst] += a * fs[src]; 32 lanes/edge, float4 gather
__global__ __launch_bounds__(256) void agg_kernel(
    const float* __restrict__ fs, const int* __restrict__ src,
    const int* __restrict__ dst, const float* __restrict__ escore,
    const float* __restrict__ denom, float* __restrict__ out, int E) {
  long t = blockIdx.x * 256L + threadIdx.x;
  long e = t >> 5;
  if (e >= E) return;
  int lane = (int)(t & 31);
  int s = src[e], d = dst[e];
  int h = lane >> 3;                              // (lane*4)/32
  float a = escore[e * HH + h] / denom[(size_t)d * HH + h];
  float4 v = *(const float4*)(fs + (size_t)s * HD + lane * 4);
  float* op = out + (size_t)d * HD + lane * 4;
  atomicAdd(op + 0, a * v.x); atomicAdd(op + 1, a * v.y);
  atomicAdd(op + 2, a * v.z); atomicAdd(op + 3, a * v.w);
}

// rst = (ft0+ft1)*0.5 (already summed in `out`); L2-normalize over D per (n,h)
__global__ __launch_bounds__(256) void norm_kernel(float* __restrict__ out, int Nn) {
  int t = blockIdx.x * 256 + threadIdx.x;
  if (t >= Nn * HH) return;
  float* p = out + (size_t)(t >> 2) * HD + (t & 3) * DD;
  float v[DD];
  float ss = 0.f;
#pragma unroll
  for (int i = 0; i < DD; ++i) { v[i] = p[i] * 0.5f; ss += v[i] * v[i]; }
  float inv = 1.0f / fmaxf(sqrtf(ss), 1e-12f);
#pragma unroll
  for (int i = 0; i < DD; ++i) p[i] = v[i] * inv;
}

// ---------------------------------------------------------------------------
extern "C" void kernel_launch(void* const* d_in, const int* in_sizes, int n_in,
                              void* d_out, int out_size, void* d_ws, size_t ws_size,
                              hipStream_t stream) {
  const float* feat = (const float*)d_in[0];
  const float* Wfc  = (const float*)d_in[1];
  const float* al   = (const float*)d_in[2];
  const float* ar   = (const float*)d_in[3];
  const float* tal  = (const float*)d_in[4];
  const float* tar  = (const float*)d_in[5];
  const int* srcs[2] = {(const int*)d_in[6], (const int*)d_in[8]};
  const int* dsts[2] = {(const int*)d_in[7], (const int*)d_in[9]};

  const int Nn = in_sizes[0] / EMB;   // 100000
  const int E  = in_sizes[6];         // 800000
  float* out = (float*)d_out;

  // ---- workspace layout (floats) : ~124 MB total -------------------------
  float*    fs     = (float*)d_ws;                      // Nn*HD
  float*    acc    = fs    + (size_t)Nn * HD;           // Nn*HD  (+deg adjacent)
  float*    deg    = acc   + (size_t)Nn * HD;           // Nn
  float*    el     = deg   + Nn;                        // Nn*H
  float*    er     = el    + (size_t)Nn * HH;           // Nn*H
  float*    gate   = er    + (size_t)Nn * HH;           // Nn*H
  unsigned* emaxE  = (unsigned*)(gate + (size_t)Nn * HH); // Nn*H
  float*    denom  = (float*)(emaxE + (size_t)Nn * HH);   // Nn*H
  float*    escore = denom + (size_t)Nn * HH;           // E*H

  // ---- phase 1: projection GEMM via fp32 WMMA ----------------------------
  int rowTiles = Nn / 16;   // 6250 (exact)
  proj_kernel<<<rowTiles, 256, 0, stream>>>(feat, Wfc, fs, rowTiles);

  // ft accumulator (ft0 + ft1) lives in d_out; zero it first
  zero_kernel<<<1024, 256, 0, stream>>>(out, (long)Nn * HD);

  const int eh_blocks  = (int)(((long)E * HH + 255) / 256);
  const int e32_blocks = (int)(((long)E * 32 + 255) / 256);
  const int nh_blocks  = (Nn * HH + 255) / 256;

  for (int et = 0; et < 2; ++et) {
    const int* src = srcs[et];
    const int* dst = dsts[et];
    // zero acc + deg (adjacent regions, one pass)
    zero_kernel<<<2048, 256, 0, stream>>>(acc, (long)Nn * (HD + 1));
    scatter_sum_kernel<<<e32_blocks, 256, 0, stream>>>(fs, src, dst, acc, deg, E);
    node_kernel<<<nh_blocks, 256, 0, stream>>>(fs, acc, deg, al, ar, tal, tar,
                                               el, er, gate, emaxE, denom, Nn);
    score_kernel<<<eh_blocks, 256, 0, stream>>>(src, dst, el, er, gate,
                                                escore, emaxE, E);
    exp_kernel<<<eh_blocks, 256, 0, stream>>>(dst, escore, emaxE, denom, E);
    agg_kernel<<<e32_blocks, 256, 0, stream>>>(fs, src, dst, escore, denom, out, E);
  }

  norm_kernel<<<nh_blocks, 256, 0, stream>>>(out, Nn);
}